// HBondWholePoseScoringModule_55559696941535
// MI455X (gfx1250) — compile-verified
//
#include <hip/hip_runtime.h>
#include <hip/hip_bf16.h>

// Problem constants (match reference)
#define Pn   8
#define Bn   160
#define Tn   32
#define MDn  8
#define MAn  8
#define NDn  6
#define NAn  6
#define Kn   11
#define MIN_SEP 4
#define NH   (Bn * MDn)   // 1280 donors (and acceptors) per pose
#define NTIL (NH / 16)    // 80 tiles per dimension
#define SJ   4            // j sub-tiles per wave (16x64 strip, 4 WMMAs)

typedef float v2f __attribute__((ext_vector_type(2)));
typedef float v8f __attribute__((ext_vector_type(8)));

// ---------------------------------------------------------------------------
// Phase 1: gather donor/acceptor staging arrays into workspace.
//   Hstage[p*NH+i] = (-2x,-2y,-2z,0)  (pre-scaled so WMMA Gram gives -2*H.A)
//   hsq   [p*NH+i] = |H|^2
//   hmeta [p*NH+i] = donor type, or -1 if slot invalid
//   (same for acceptors, unscaled coords)
// Also zeroes d_out (harness poisons it before timing).
// ---------------------------------------------------------------------------
__global__ void hbond_prep(const float* __restrict__ coords,
                           const int*   __restrict__ block_type,
                           const int*   __restrict__ n_donH,
                           const int*   __restrict__ donH_inds,
                           const int*   __restrict__ donH_type,
                           const int*   __restrict__ n_acc,
                           const int*   __restrict__ acc_inds,
                           const int*   __restrict__ acc_type,
                           float4* __restrict__ Hstage, float* __restrict__ hsq, int* __restrict__ hmeta,
                           float4* __restrict__ Astage, float* __restrict__ asq, int* __restrict__ ameta,
                           float* __restrict__ out)
{
    int gid = blockIdx.x * blockDim.x + threadIdx.x;
    if (gid < Pn) out[gid] = 0.0f;
    if (gid >= Pn * NH) return;

    int p = gid / NH;
    int i = gid % NH;
    int b = i >> 3;        // block index
    int m = i & 7;         // slot within block
    int bt = block_type[p * Bn + b];

    // donor hydrogen
    {
        int t = donH_inds[bt * MDn + m];
        const float* c = &coords[((size_t)(p * Bn + b) * Tn + t) * 3];
        float x = c[0], y = c[1], z = c[2];
        Hstage[gid] = make_float4(-2.0f * x, -2.0f * y, -2.0f * z, 0.0f);
        hsq[gid]    = x * x + y * y + z * z;
        hmeta[gid]  = (m < n_donH[bt]) ? donH_type[bt * MDn + m] : -1;
    }
    // acceptor
    {
        int t = acc_inds[bt * MAn + m];
        const float* c = &coords[((size_t)(p * Bn + b) * Tn + t) * 3];
        float x = c[0], y = c[1], z = c[2];
        Astage[gid] = make_float4(x, y, z, 0.0f);
        asq[gid]    = x * x + y * y + z * z;
        ameta[gid]  = (m < n_acc[bt]) ? acc_type[bt * MAn + m] : -1;
    }
}

// ---------------------------------------------------------------------------
// Phase 2: one wave32 per 16x64 strip (1 donor tile x 4 acceptor tiles) of
// the (1280 x 1280) interaction matrix.  -2*H.A Gram term via four
// V_WMMA_F32_16X16X4_F32 sharing the A operand, then per-element sqrt +
// type gate + Horner polynomial, wave reduction, atomic accumulate.
// Grid: (NTIL/SJ = 20, NTIL/8 = 10, Pn = 8), 256 threads = 8 waves/block.
// ---------------------------------------------------------------------------
__global__ void __launch_bounds__(256)
hbond_main(const float* __restrict__ pair_params,      // (ND,NA,3)
           const float* __restrict__ pair_poly,        // (ND,NA,K)
           const float* __restrict__ gparams,          // (1,3)
           const int*   __restrict__ sep,              // (P,B,B)
           const float4* __restrict__ Hstage, const float* __restrict__ hsq, const int* __restrict__ hmeta,
           const float4* __restrict__ Astage, const float* __restrict__ asq, const int* __restrict__ ameta,
           float* __restrict__ out)
{
    __shared__ float s_pp[NDn * NAn * 3];
    __shared__ float s_poly[NDn * NAn * Kn];
    __shared__ float s_g;

    for (int t = threadIdx.x; t < NDn * NAn * 3;  t += blockDim.x) s_pp[t]   = pair_params[t];
    for (int t = threadIdx.x; t < NDn * NAn * Kn; t += blockDim.x) s_poly[t] = pair_poly[t];
    if (threadIdx.x == 0) s_g = gparams[0];
    __syncthreads();

    const int wave = threadIdx.x >> 5;
    const int lane = threadIdx.x & 31;

    const int p   = blockIdx.z;
    const int ti  = blockIdx.y * 8 + wave;   // donor tile row
    const int tj0 = blockIdx.x * SJ;         // first acceptor tile column

    const int hi = lane >> 4;     // 0 or 1 (lane half -> K pair select)
    const int lo = lane & 15;

    const int pbase = p * NH;

    // A operand: donor row M = lo ; K pair = (2*hi, 2*hi+1)
    const float4 hv = Hstage[pbase + ti * 16 + lo];
    v2f a;
    a.x = hi ? hv.z : hv.x;  a.y = hi ? hv.w : hv.y;

    // Four B operands (acceptor columns) + four Gram WMMAs, all convergent.
    v8f   c[SJ];
    float asq_s[SJ];
    int   at_s[SJ];
    int   b2_s[SJ];
#pragma unroll
    for (int s = 0; s < SJ; ++s) {
        const int j = (tj0 + s) * 16 + lo;
        const float4 av = Astage[pbase + j];
        v2f b;
        b.x = hi ? av.z : av.x;  b.y = hi ? av.w : av.y;
        v8f z = {};
        c[s] = __builtin_amdgcn_wmma_f32_16x16x4_f32(false, a, false, b,
                                                     (short)0, z, false, false);
        asq_s[s] = asq[pbase + j];
        at_s[s]  = ameta[pbase + j];
        b2_s[s]  = j >> 3;
    }

    float sum = 0.0f;
#pragma unroll
    for (int r = 0; r < 8; ++r) {
        const int   i     = ti * 16 + r + hi * 8;    // donor row for this VGPR
        const float hsq_i = hsq[pbase + i];
        const int   dt    = hmeta[pbase + i];
        const int   b1    = i >> 3;
        const int   srow  = (p * Bn + b1) * Bn;

#pragma unroll
        for (int s = 0; s < SJ; ++s) {
            const float d2 = c[s][r] + hsq_i + asq_s[s];
            const float d  = __builtin_amdgcn_sqrtf(fmaxf(d2, 0.0f) + 1e-12f);

            if ((dt >= 0) & (at_s[s] >= 0) & (b1 != b2_s[s])) {
                if (sep[srow + b2_s[s]] >= MIN_SEP) {
                    const int pi = dt * NAn + at_s[s];
                    const float dmin = s_pp[pi * 3 + 0];
                    const float dmax = s_pp[pi * 3 + 1];
                    if (d >= dmin && d <= dmax) {
                        const float w = s_pp[pi * 3 + 2];
                        const float* q = &s_poly[pi * Kn];
                        float E = q[0];
#pragma unroll
                        for (int k = 1; k < Kn; ++k) E = E * d + q[k];
                        sum += w * E * s_g;
                    }
                }
            }
        }
    }

    // wave32 reduction
#pragma unroll
    for (int off = 16; off > 0; off >>= 1)
        sum += __shfl_xor(sum, off, 32);

    if (lane == 0)
        atomicAdd(&out[p], sum);
}

// ---------------------------------------------------------------------------
extern "C" void kernel_launch(void* const* d_in, const int* in_sizes, int n_in,
                              void* d_out, int out_size, void* d_ws, size_t ws_size,
                              hipStream_t stream)
{
    const float* coords      = (const float*)d_in[0];
    const float* pair_params = (const float*)d_in[1];
    const float* pair_poly   = (const float*)d_in[2];
    const float* gparams     = (const float*)d_in[3];
    const int*   block_type  = (const int*)d_in[4];
    const int*   sep         = (const int*)d_in[5];
    const int*   n_donH      = (const int*)d_in[6];
    const int*   donH_inds   = (const int*)d_in[7];
    const int*   donH_type   = (const int*)d_in[8];
    const int*   n_acc       = (const int*)d_in[9];
    const int*   acc_inds    = (const int*)d_in[10];
    const int*   acc_type    = (const int*)d_in[11];

    char* ws = (char*)d_ws;
    const size_t nstage = (size_t)Pn * NH;           // 10240 entries
    float4* Hstage = (float4*)(ws);                              // 160 KB
    float4* Astage = (float4*)(ws + nstage * 16);                // 160 KB
    float*  hsq    = (float*) (ws + nstage * 32);                // 40 KB
    float*  asq    = (float*) (ws + nstage * 36);                // 40 KB
    int*    hmeta  = (int*)   (ws + nstage * 40);                // 40 KB
    int*    ameta  = (int*)   (ws + nstage * 44);                // 40 KB

    float* out = (float*)d_out;

    const int prepThreads = Pn * NH;                 // 10240
    hbond_prep<<<(prepThreads + 255) / 256, 256, 0, stream>>>(
        coords, block_type, n_donH, donH_inds, donH_type,
        n_acc, acc_inds, acc_type,
        Hstage, hsq, hmeta, Astage, asq, ameta, out);

    dim3 grid(NTIL / SJ, NTIL / 8, Pn);              // (20, 10, 8)
    hbond_main<<<grid, 256, 0, stream>>>(
        pair_params, pair_poly, gparams, sep,
        Hstage, hsq, hmeta, Astage, asq, ameta, out);
}